// EdgeMaskCounterfactual_70806830842664
// MI455X (gfx1250) — compile-verified
//
#include <hip/hip_runtime.h>
#include <math.h>

// ---------------------------------------------------------------------------
// CDNA5 (gfx1250) fp32 WMMA helpers: V_WMMA_F32_16X16X4_F32
//   A: 16x4 f32  -> 2 VGPRs/lane (v2f), lane holds A[lane&15][2*(lane>>4)+v]
//   B: 4x16 f32  -> 2 VGPRs/lane (v2f), lane holds B[2*(lane>>4)+v][lane&15]
//   C/D: 16x16   -> 8 VGPRs/lane (v8f), c[v] = C[v + 8*(lane>>4)][lane&15]
// ---------------------------------------------------------------------------
typedef __attribute__((ext_vector_type(2))) float v2f;
typedef __attribute__((ext_vector_type(8))) float v8f;

__device__ __forceinline__ v8f wmma_f32_16x16x4(v2f a, v2f b, v8f c) {
  // 8 args: (neg_a, A, neg_b, B, c_mod, C, reuse_a, reuse_b)
  return __builtin_amdgcn_wmma_f32_16x16x4_f32(false, a, false, b, (short)0, c,
                                               false, false);
}

__device__ __forceinline__ void gatomic_add(float* p, float v) {
  __hip_atomic_fetch_add(p, v, __ATOMIC_RELAXED, __HIP_MEMORY_SCOPE_AGENT);
}

// Async global -> LDS copy of 16 bytes (CDNA5 GLOBAL_LOAD_ASYNC_TO_LDS_B128,
// tracked on ASYNCcnt).  lds_off is the wave-relative LDS byte address
// (low 32 bits of the flat shared-memory address per ISA 10.2).
__device__ __forceinline__ void async_copy_b128(unsigned lds_off,
                                                const void* gptr) {
  asm volatile("global_load_async_to_lds_b128 %0, %1, off" ::"v"(lds_off),
               "v"((unsigned long long)gptr)
               : "memory");
}
__device__ __forceinline__ void async_wait0() {
  asm volatile("s_wait_asynccnt 0x0" ::: "memory");
}

// ---------------------------------------------------------------------------
// Elementwise utility kernels
// ---------------------------------------------------------------------------
__global__ void k_fill(float* __restrict__ p, float v, int n) {
  int i = blockIdx.x * blockDim.x + threadIdx.x;
  if (i < n) p[i] = v;
}

__global__ void k_relu(float* __restrict__ p, int n) {
  int i = blockIdx.x * blockDim.x + threadIdx.x;
  if (i < n) p[i] = fmaxf(p[i], 0.0f);
}

__global__ void k_mask_scatter(const int* __restrict__ sub_idx,
                               const float* __restrict__ m_logits,
                               float* __restrict__ mask, int S) {
  int i = blockIdx.x * blockDim.x + threadIdx.x;
  if (i < S) mask[sub_idx[i]] = 1.0f / (1.0f + __expf(-m_logits[i]));
}

// agg[row,:] *= 1/max(cnt[row],1)   (K = 256, so row = i >> 8)
__global__ void k_rownorm(float* __restrict__ agg, const float* __restrict__ cnt,
                          int total) {
  int i = blockIdx.x * blockDim.x + threadIdx.x;
  if (i < total) agg[i] *= 1.0f / fmaxf(cnt[i >> 8], 1.0f);
}

// ---------------------------------------------------------------------------
// Per-relation aggregation: atomic scatter of x[src] rows into agg[dst],
// plus edge counts.  One block (256 threads) per edge; K == 256.
// ---------------------------------------------------------------------------
__global__ __launch_bounds__(256) void k_scatter_rel(
    const float* __restrict__ x, const int* __restrict__ src,
    const int* __restrict__ dst, const int* __restrict__ etype,
    float* __restrict__ agg, float* __restrict__ cnt, int E, int K, int rel) {
  int e = blockIdx.x;
  if (e >= E) return;
  if (etype[e] != rel) return;
  int s = src[e];
  int d = dst[e];
  const float* xr = x + (size_t)s * K;
  float* ar = agg + (size_t)d * K;
  for (int c = threadIdx.x; c < K; c += blockDim.x) gatomic_add(&ar[c], xr[c]);
  if (threadIdx.x == 0) gatomic_add(&cnt[d], 1.0f);
}

// ---------------------------------------------------------------------------
// WMMA fp32 GEMM, 4 M-tiles per wave so each B (weight) fragment is reused
// by 4 WMMAs (weights are L2-resident; A rows stream from HBM).
//   ACC == false: Out = X @ W + bias     (accumulators start at bias)
//   ACC == true : Out += X @ W           (accumulators start at Out)
// grid = (ceil(Mtiles/4), ceil((NO/16)/8)), block = 256 (8 waves).
// ---------------------------------------------------------------------------
template <bool ACC>
__global__ __launch_bounds__(256) void k_gemm(
    const float* __restrict__ X, const float* __restrict__ W,
    const float* __restrict__ bias, float* __restrict__ Out, int M, int K,
    int NO) {
  int wave = threadIdx.x >> 5;
  int lane = threadIdx.x & 31;
  int m = lane & 15;
  int hi = lane >> 4;
  int ntile = blockIdx.y * 8 + wave;
  int ntiles = NO >> 4;
  if (ntile >= ntiles) return;  // wave-uniform: EXEC all-1s for WMMA
  int n = ntile * 16 + m;       // B / Out column this lane touches
  int Mtiles = M >> 4;
  int mt0 = blockIdx.x * 4;

  v8f acc[4];
  const float* xr[4];
  bool valid[4];
#pragma unroll
  for (int t = 0; t < 4; ++t) {
    int mtile = mt0 + t;
    valid[t] = (mtile < Mtiles);
    int mtc = valid[t] ? mtile : (Mtiles - 1);  // clamp: safe addr, no store
    xr[t] = X + (size_t)(mtc * 16 + m) * K;
    __builtin_prefetch(xr[t], 0, 3);  // global_prefetch_b8 on the A stream
    if (ACC) {
#pragma unroll
      for (int v = 0; v < 8; ++v)
        acc[t][v] = Out[(size_t)(mtc * 16 + v + 8 * hi) * NO + n];
    } else {
      float bv = bias[n];
#pragma unroll
      for (int v = 0; v < 8; ++v) acc[t][v] = bv;
    }
  }

  for (int kk = 0; kk < K; kk += 4) {
    int k0 = kk + 2 * hi;
    v2f b;
    b.x = W[(size_t)k0 * NO + n];
    b.y = W[(size_t)(k0 + 1) * NO + n];
#pragma unroll
    for (int t = 0; t < 4; ++t) {
      v2f a;
      a.x = xr[t][k0];
      a.y = xr[t][k0 + 1];
      acc[t] = wmma_f32_16x16x4(a, b, acc[t]);
    }
  }

#pragma unroll
  for (int t = 0; t < 4; ++t) {
    if (valid[t]) {
#pragma unroll
      for (int v = 0; v < 8; ++v)
        Out[(size_t)((mt0 + t) * 16 + v + 8 * hi) * NO + n] = acc[t][v];
    }
  }
}

// ---------------------------------------------------------------------------
// Fused edge MLP: per block of 16 edges, stage A[16,512] =
// [h2[src] | h2[dst] | mask*e_text] into LDS with async global->LDS b128
// copies (ASYNCcnt), then 16 column tiles of [16,512]x[512,256] via WMMA,
// fusing bias+ReLU+dot(Wm2)+bias into logits.
// LDS stride 516 floats: 16 row-parallel fragment reads hit distinct banks,
// and each 4-float chunk stays 16B-aligned (516*4 = 2064 = 16*129).
// ---------------------------------------------------------------------------
#define EM_LDA 516
__global__ __launch_bounds__(256) void k_edge_mlp(
    const float* __restrict__ h2,     // [N,128]
    const int* __restrict__ src, const int* __restrict__ dst,
    const float* __restrict__ etext,  // [E,256]
    const float* __restrict__ mask,   // [E]
    const float* __restrict__ Wm1,    // [512,256]
    const float* __restrict__ bm1,    // [256]
    const float* __restrict__ Wm2,    // [256]
    const float* __restrict__ bm2,    // [1]
    float* __restrict__ logits, int E) {
  __shared__ float At[16 * EM_LDA];
  __shared__ float accrow[16];
  int tid = threadIdx.x;
  int tile = blockIdx.x;

  // ---- async stage of the gathered/concatenated A tile (2048 b128 chunks) --
  for (int i = tid; i < 2048; i += 256) {
    int r = i >> 7;             // edge row 0..15
    int c4 = (i & 127) << 2;    // feature col 0..508, step 4
    int e = tile * 16 + r;
    const float* g;
    if (c4 < 128)
      g = h2 + (size_t)src[e] * 128 + c4;
    else if (c4 < 256)
      g = h2 + (size_t)dst[e] * 128 + (c4 - 128);
    else
      g = etext + (size_t)e * 256 + (c4 - 256);
    unsigned lds_off = (unsigned)(uintptr_t)&At[r * EM_LDA + c4];
    async_copy_b128(lds_off, g);
  }
  async_wait0();    // this wave's async copies landed in LDS
  __syncthreads();  // all waves' copies visible

  // scale the text region by the edge mask, in LDS
  for (int i = tid; i < 16 * 256; i += 256) {
    int r = i >> 8;
    At[r * EM_LDA + 256 + (i & 255)] *= mask[tile * 16 + r];
  }
  if (tid < 16) accrow[tid] = 0.0f;
  __syncthreads();

  int wave = tid >> 5;
  int lane = tid & 31;
  int m = lane & 15;
  int hi = lane >> 4;

  float pacc[8];
#pragma unroll
  for (int v = 0; v < 8; ++v) pacc[v] = 0.0f;

  for (int nt = wave; nt < 16; nt += 8) {  // 2 column tiles per wave
    int n = nt * 16 + m;
    v8f c;
#pragma unroll
    for (int v = 0; v < 8; ++v) c[v] = 0.0f;
    for (int kk = 0; kk < 512; kk += 4) {
      int k0 = kk + 2 * hi;
      v2f a;
      a.x = At[m * EM_LDA + k0];
      a.y = At[m * EM_LDA + k0 + 1];
      v2f b;
      b.x = Wm1[(size_t)k0 * 256 + n];
      b.y = Wm1[(size_t)(k0 + 1) * 256 + n];
      c = wmma_f32_16x16x4(a, b, c);
    }
    float bb = bm1[n];
    float w2 = Wm2[n];
#pragma unroll
    for (int v = 0; v < 8; ++v) pacc[v] += fmaxf(c[v] + bb, 0.0f) * w2;
  }

  // one cross-lane reduction for both column tiles: sum over the 16 lanes
  // sharing output row (v + 8*hi), then one LDS atomic per row per wave.
#pragma unroll
  for (int v = 0; v < 8; ++v) {
    float p = pacc[v];
    p += __shfl_xor(p, 1, 32);
    p += __shfl_xor(p, 2, 32);
    p += __shfl_xor(p, 4, 32);
    p += __shfl_xor(p, 8, 32);
    if (m == 0) atomicAdd(&accrow[v + 8 * hi], p);
  }
  __syncthreads();
  if (tid < 16) logits[tile * 16 + tid] = accrow[tid] + bm2[0];
}

// ---------------------------------------------------------------------------
// Launcher
// ---------------------------------------------------------------------------
extern "C" void kernel_launch(void* const* d_in, const int* in_sizes, int n_in,
                              void* d_out, int out_size, void* d_ws,
                              size_t ws_size, hipStream_t stream) {
  (void)in_sizes; (void)n_in; (void)out_size; (void)ws_size;
  const int N = 50000, E = 400000, R = 8, S = 1024;

  const int* edge_index = (const int*)d_in[0];
  const int* src = edge_index;
  const int* dst = edge_index + E;
  const int* etype = (const int*)d_in[1];
  const float* etext = (const float*)d_in[2];
  const int* sub_idx = (const int*)d_in[3];
  const float* m_logits = (const float*)d_in[4];
  const float* node_emb = (const float*)d_in[5];
  const float* W1 = (const float*)d_in[6];      // [8,256,256]
  const float* root1 = (const float*)d_in[7];   // [256,256]
  const float* b1 = (const float*)d_in[8];
  const float* W2 = (const float*)d_in[9];      // [8,256,128]
  const float* root2 = (const float*)d_in[10];  // [256,128]
  const float* b2 = (const float*)d_in[11];
  const float* Wm1 = (const float*)d_in[12];    // [512,256]
  const float* bm1 = (const float*)d_in[13];
  const float* Wm2 = (const float*)d_in[14];    // [256]
  const float* bm2 = (const float*)d_in[15];

  float* logits = (float*)d_out;
  float* mask = logits + E;

  // workspace: bufA [N,256] (out1 -> h1), bufB [N,256] (agg), bufC [N,128]
  // (out2 -> h2), cnt [N]  => ~128.2 MB
  float* bufA = (float*)d_ws;
  float* bufB = bufA + (size_t)N * 256;
  float* bufC = bufB + (size_t)N * 256;
  float* cnt = bufC + (size_t)N * 128;

  // ---- edge mask -----------------------------------------------------------
  k_fill<<<(E + 255) / 256, 256, 0, stream>>>(mask, 1.0f, E);
  k_mask_scatter<<<(S + 255) / 256, 256, 0, stream>>>(sub_idx, m_logits, mask, S);

  const int Mtiles = N / 16;             // 3125 (exact)
  const int Mblk = (Mtiles + 3) / 4;     // 4 M-tiles per wave

  // ---- RGCN layer 1: bufA = relu(node_emb@root1 + b1 + sum_r mean_r@W1[r]) --
  k_gemm<false><<<dim3(Mblk, 2), 256, 0, stream>>>(node_emb, root1, b1, bufA,
                                                   N, 256, 256);
  for (int r = 0; r < R; ++r) {
    k_fill<<<(N * 256 + 255) / 256, 256, 0, stream>>>(bufB, 0.0f, N * 256);
    k_fill<<<(N + 255) / 256, 256, 0, stream>>>(cnt, 0.0f, N);
    k_scatter_rel<<<E, 256, 0, stream>>>(node_emb, src, dst, etype, bufB, cnt,
                                         E, 256, r);
    k_rownorm<<<(N * 256 + 255) / 256, 256, 0, stream>>>(bufB, cnt, N * 256);
    k_gemm<true><<<dim3(Mblk, 2), 256, 0, stream>>>(
        bufB, W1 + (size_t)r * 256 * 256, nullptr, bufA, N, 256, 256);
  }
  k_relu<<<(N * 256 + 255) / 256, 256, 0, stream>>>(bufA, N * 256);

  // ---- RGCN layer 2: bufC = relu(h1@root2 + b2 + sum_r mean_r@W2[r]) -------
  k_gemm<false><<<dim3(Mblk, 1), 256, 0, stream>>>(bufA, root2, b2, bufC, N,
                                                   256, 128);
  for (int r = 0; r < R; ++r) {
    k_fill<<<(N * 256 + 255) / 256, 256, 0, stream>>>(bufB, 0.0f, N * 256);
    k_fill<<<(N + 255) / 256, 256, 0, stream>>>(cnt, 0.0f, N);
    k_scatter_rel<<<E, 256, 0, stream>>>(bufA, src, dst, etype, bufB, cnt, E,
                                         256, r);
    k_rownorm<<<(N * 256 + 255) / 256, 256, 0, stream>>>(bufB, cnt, N * 256);
    k_gemm<true><<<dim3(Mblk, 1), 256, 0, stream>>>(
        bufB, W2 + (size_t)r * 256 * 128, nullptr, bufC, N, 256, 128);
  }
  k_relu<<<(N * 128 + 255) / 256, 256, 0, stream>>>(bufC, N * 128);

  // ---- fused edge MLP ------------------------------------------------------
  k_edge_mlp<<<E / 16, 256, 0, stream>>>(bufC, src, dst, etext, mask, Wm1, bm1,
                                         Wm2, bm2, logits, E);
}